// MultimodalFusion_7112465842292
// MI455X (gfx1250) — compile-verified
//
#include <hip/hip_runtime.h>
#include <cstddef>

#define B_ 256
#define S_ 512
#define H_ 768
#define D_ 2048

typedef __attribute__((ext_vector_type(2))) float v2f;
typedef __attribute__((ext_vector_type(4))) float v4f;
typedef __attribute__((ext_vector_type(8))) float v8f;

// ============ WMMA fp32 GEMM: C[M,N] = alpha*(A@B) (+ bias) ============
// One wave per 16x16 C tile. A: MxK row-major. B: KxN row-major (BT=false)
// or B stored NxK row-major meaning logical B[k,n] = Bm[n*K+k] (BT=true).
template<bool BT>
__global__ __launch_bounds__(32) void mm_gemm_wmma(const float* __restrict__ A,
                                                   const float* __restrict__ Bm,
                                                   const float* __restrict__ bias,
                                                   float* __restrict__ C,
                                                   int M, int N, int K, float alpha)
{
    const int lane  = threadIdx.x;
    const int nBase = blockIdx.x * 16;
    const int mBase = blockIdx.y * 16;
    const int r16   = lane & 15;           // A row / B col within tile
    const int kq    = (lane >> 4) << 1;    // lanes 0-15: k={0,1}, lanes16-31: k={2,3}

    v8f acc = {};
    const float* Arow  = A  + (size_t)(mBase + r16) * K + kq;
    const float* BrowT = Bm + (size_t)(nBase + r16) * K + kq;

    for (int k0 = 0; k0 < K; k0 += 4) {
        v2f a = *(const v2f*)(Arow + k0);
        v2f b;
        if (BT) {
            b = *(const v2f*)(BrowT + k0);
        } else {
            b.x = Bm[(size_t)(k0 + kq)     * N + nBase + r16];
            b.y = Bm[(size_t)(k0 + kq + 1) * N + nBase + r16];
        }
        acc = __builtin_amdgcn_wmma_f32_16x16x4_f32(false, a, false, b,
                                                    (short)0, acc, false, false);
    }
    const int mOff = (lane >> 4) * 8;      // C/D layout: lanes16-31 hold M=r+8
    #pragma unroll
    for (int r = 0; r < 8; ++r) {
        int mm = mBase + mOff + r;
        int nn = nBase + r16;
        float v = acc[r] * alpha;
        if (bias) v += bias[nn];
        C[(size_t)mm * N + nn] = v;
    }
}

// ============ sbias[b] = scale * dot(img_q[b], btk) ============
__global__ __launch_bounds__(256) void mm_sbias(const float* __restrict__ q,
                                                const float* __restrict__ btk,
                                                float* __restrict__ sbias, float scale)
{
    __shared__ float red[256];
    const int b = blockIdx.x, tid = threadIdx.x;
    float s = 0.f;
    #pragma unroll
    for (int k = 0; k < 3; ++k) {
        int c = tid + k * 256;
        s += q[(size_t)b * H_ + c] * btk[c];
    }
    red[tid] = s; __syncthreads();
    for (int off = 128; off > 0; off >>= 1) {
        if (tid < off) red[tid] += red[tid + off];
        __syncthreads();
    }
    if (tid == 0) sbias[b] = red[0] * scale;
}

// ============ single-pass streaming attention pool over text_features ========
// ctx[b,:] = softmax_s( qk_b . x_{b,s} + sbias_b ) @ x_b      (online softmax)
__global__ __launch_bounds__(512) void mm_attn_stream(const float* __restrict__ text,
                                                      const float* __restrict__ qk,
                                                      const float* __restrict__ sbias,
                                                      float* __restrict__ ctx)
{
    __shared__ float s_acc[16 * H_];
    __shared__ float s_m[16], s_l[16];

    const int b    = blockIdx.x;
    const int tid  = threadIdx.x;
    const int wave = tid >> 5;
    const int lane = tid & 31;
    const float sb = sbias[b];

    v4f q[6];
    const v4f* qrow = (const v4f*)(qk + (size_t)b * H_);
    #pragma unroll
    for (int j = 0; j < 6; ++j) q[j] = qrow[j * 32 + lane];

    v4f acc[6];
    #pragma unroll
    for (int j = 0; j < 6; ++j) acc[j] = (v4f){0.f, 0.f, 0.f, 0.f};
    float m = -3.0e38f, l = 0.f;

    const v4f* trow0 = (const v4f*)(text + (size_t)b * S_ * H_);
    for (int i = 0; i < 32; ++i) {
        const int s = wave * 32 + i;
        const v4f* xr = trow0 + (size_t)s * (H_ / 4);
        v4f x[6];
        #pragma unroll
        for (int j = 0; j < 6; ++j) x[j] = xr[j * 32 + lane];
        float d = 0.f;
        #pragma unroll
        for (int j = 0; j < 6; ++j)
            d += x[j].x * q[j].x + x[j].y * q[j].y + x[j].z * q[j].z + x[j].w * q[j].w;
        #pragma unroll
        for (int off = 16; off > 0; off >>= 1) d += __shfl_xor(d, off, 32);

        const float sc = d + sb;
        const float mn = fmaxf(m, sc);
        const float f  = __expf(m - mn);
        const float e  = __expf(sc - mn);
        l = l * f + e;
        #pragma unroll
        for (int j = 0; j < 6; ++j) acc[j] = acc[j] * f + x[j] * e;
        m = mn;
    }

    #pragma unroll
    for (int j = 0; j < 6; ++j)
        *(v4f*)&s_acc[wave * H_ + (j * 32 + lane) * 4] = acc[j];
    if (lane == 0) { s_m[wave] = m; s_l[wave] = l; }
    __syncthreads();

    float M = -3.0e38f;
    #pragma unroll
    for (int w = 0; w < 16; ++w) M = fmaxf(M, s_m[w]);
    float wexp[16]; float L = 0.f;
    #pragma unroll
    for (int w = 0; w < 16; ++w) { wexp[w] = __expf(s_m[w] - M); L += wexp[w] * s_l[w]; }
    const float invL = 1.0f / L;
    for (int c = tid; c < H_; c += 512) {
        float sum = 0.f;
        #pragma unroll
        for (int w = 0; w < 16; ++w) sum += wexp[w] * s_acc[w * H_ + c];
        ctx[(size_t)b * H_ + c] = sum * invL;
    }
}

// ============ LayerNorm (+optional residual, +optional exact GELU) ============
__global__ __launch_bounds__(256) void mm_ln(const float* __restrict__ x,
                                             const float* __restrict__ res, long res_stride,
                                             const float* __restrict__ g,
                                             const float* __restrict__ bb,
                                             float* __restrict__ out, int do_gelu)
{
    __shared__ float red[256];
    const int b = blockIdx.x, tid = threadIdx.x;
    const float* xr = x + (size_t)b * H_;
    const float* rr = res ? res + (size_t)b * res_stride : nullptr;
    float v[3]; float s = 0.f;
    #pragma unroll
    for (int k = 0; k < 3; ++k) {
        int c = tid + k * 256;
        float t = xr[c];
        if (rr) t += rr[c];
        v[k] = t; s += t;
    }
    red[tid] = s; __syncthreads();
    for (int off = 128; off > 0; off >>= 1) { if (tid < off) red[tid] += red[tid + off]; __syncthreads(); }
    const float mu = red[0] * (1.0f / H_);
    __syncthreads();
    float s2 = 0.f;
    #pragma unroll
    for (int k = 0; k < 3; ++k) { float d = v[k] - mu; s2 += d * d; }
    red[tid] = s2; __syncthreads();
    for (int off = 128; off > 0; off >>= 1) { if (tid < off) red[tid] += red[tid + off]; __syncthreads(); }
    const float inv = rsqrtf(red[0] * (1.0f / H_) + 1e-5f);
    #pragma unroll
    for (int k = 0; k < 3; ++k) {
        int c = tid + k * 256;
        float y = (v[k] - mu) * inv * g[c] + bb[c];
        if (do_gelu) y = 0.5f * y * (1.0f + erff(y * 0.70710678118654752f));
        out[(size_t)b * H_ + c] = y;
    }
}

// ============ gate = softmax([tc,ic]@Wg + bg); weighted = [tc*g0, ic*g1] =====
__global__ __launch_bounds__(256) void mm_gate(const float* __restrict__ tc,
                                               const float* __restrict__ ic,
                                               const float* __restrict__ Wg,
                                               const float* __restrict__ bg,
                                               float* __restrict__ wgt)
{
    __shared__ float r0[256], r1[256];
    const int b = blockIdx.x, tid = threadIdx.x;
    float p0 = 0.f, p1 = 0.f;
    for (int r = tid; r < 2 * H_; r += 256) {
        float val = (r < H_) ? tc[(size_t)b * H_ + r] : ic[(size_t)b * H_ + (r - H_)];
        p0 += val * Wg[2 * r];
        p1 += val * Wg[2 * r + 1];
    }
    r0[tid] = p0; r1[tid] = p1; __syncthreads();
    for (int off = 128; off > 0; off >>= 1) {
        if (tid < off) { r0[tid] += r0[tid + off]; r1[tid] += r1[tid + off]; }
        __syncthreads();
    }
    const float l0 = r0[0] + bg[0], l1 = r1[0] + bg[1];
    const float mx = fmaxf(l0, l1);
    const float e0 = __expf(l0 - mx), e1 = __expf(l1 - mx);
    const float inv = 1.0f / (e0 + e1);
    const float g0 = e0 * inv, g1 = e1 * inv;
    for (int c = tid; c < H_; c += 256) {
        wgt[(size_t)b * 2 * H_ + c]      = tc[(size_t)b * H_ + c] * g0;
        wgt[(size_t)b * 2 * H_ + H_ + c] = ic[(size_t)b * H_ + c] * g1;
    }
}

extern "C" void kernel_launch(void* const* d_in, const int* in_sizes, int n_in,
                              void* d_out, int out_size, void* d_ws, size_t ws_size,
                              hipStream_t stream) {
    const float* text = (const float*)d_in[0];   // (B,S,H)
    const float* img  = (const float*)d_in[1];   // (B,2048)
    const float* Wp   = (const float*)d_in[2];
    const float* bp   = (const float*)d_in[3];
    // d_in[4] Wtq, d_in[5] btq  : dead (softmax over size-1 axis == 1)
    // d_in[6] Wik, d_in[7] bik  : dead
    const float* Wiv  = (const float*)d_in[8];
    const float* biv  = (const float*)d_in[9];
    const float* Wiq  = (const float*)d_in[10];
    const float* biq  = (const float*)d_in[11];
    const float* Wtk  = (const float*)d_in[12];
    const float* btk  = (const float*)d_in[13];
    const float* Wtv  = (const float*)d_in[14];
    const float* btv  = (const float*)d_in[15];
    const float* tn_g = (const float*)d_in[16];
    const float* tn_b = (const float*)d_in[17];
    const float* in_g = (const float*)d_in[18];
    const float* in_b = (const float*)d_in[19];
    const float* Wf   = (const float*)d_in[20];
    const float* bf   = (const float*)d_in[21];
    const float* fn_g = (const float*)d_in[22];
    const float* fn_b = (const float*)d_in[23];
    const float* Wg   = (const float*)d_in[24];
    const float* bg   = (const float*)d_in[25];

    const size_t BH = (size_t)B_ * H_;
    float* ws    = (float*)d_ws;
    float* P     = ws;            // img_projected  (B,H)
    float* imgq  = P    + BH;     // (B,H)
    float* qk    = imgq + BH;     // (B,H)  scaled
    float* imgv  = qk   + BH;     // (B,H)
    float* ctx   = imgv + BH;     // attn-pooled text rows (B,H)
    float* icp   = ctx  + BH;     // ctx @ Wtv + btv
    float* tc    = icp  + BH;     // text_context
    float* ic    = tc   + BH;     // img_context
    float* opre  = ic   + BH;     // weighted @ Wf + bf
    float* wgt   = opre + BH;     // (B, 2H)
    float* sbias = wgt  + 2 * BH; // (B)

    const float scale = 0.03608439182435161f;   // 1/sqrt(768)
    dim3 blk32(32), blk256(256), blk512(512);
    dim3 gH(H_ / 16, B_ / 16);

    // img_projected = img @ Wp + bp                    (K=2048)
    mm_gemm_wmma<false><<<gH, blk32, 0, stream>>>(img, Wp, bp, P, B_, H_, D_, 1.0f);
    // img_q = P @ Wiq + biq
    mm_gemm_wmma<false><<<gH, blk32, 0, stream>>>(P, Wiq, biq, imgq, B_, H_, H_, 1.0f);
    // qk = scale * (img_q @ Wtk^T)   (Wtk is HxH row-major -> BT access)
    mm_gemm_wmma<true ><<<gH, blk32, 0, stream>>>(imgq, Wtk, nullptr, qk, B_, H_, H_, scale);
    // sbias_b = scale * img_q_b . btk
    mm_sbias<<<B_, blk256, 0, stream>>>(imgq, btk, sbias, scale);
    // img_v = P @ Wiv + biv
    mm_gemm_wmma<false><<<gH, blk32, 0, stream>>>(P, Wiv, biv, imgv, B_, H_, H_, 1.0f);
    // ctx = softmax(qk.x + sbias) @ x   -- single pass over 402 MB
    mm_attn_stream<<<B_, blk512, 0, stream>>>(text, qk, sbias, ctx);
    // img_context_pre = ctx @ Wtv + btv
    mm_gemm_wmma<false><<<gH, blk32, 0, stream>>>(ctx, Wtv, btv, icp, B_, H_, H_, 1.0f);
    // text_context = LN(img_v + cls_token); cls row = text[b, 0, :]
    mm_ln<<<B_, blk256, 0, stream>>>(imgv, text, (long)S_ * H_, tn_g, tn_b, tc, 0);
    // img_context = LN(icp + img_projected)
    mm_ln<<<B_, blk256, 0, stream>>>(icp, P, (long)H_, in_g, in_b, ic, 0);
    // gate + weighted concat
    mm_gate<<<B_, blk256, 0, stream>>>(tc, ic, Wg, bg, wgt);
    // out_pre = weighted @ Wf + bf                     (K=1536)
    mm_gemm_wmma<false><<<gH, blk32, 0, stream>>>(wgt, Wf, bf, opre, B_, H_, 2 * H_, 1.0f);
    // out = gelu(LN(out_pre), exact)
    mm_ln<<<B_, blk256, 0, stream>>>(opre, nullptr, 0, fn_g, fn_b, (float*)d_out, 1);
}